// EmotionModel_13426067767444
// MI455X (gfx1250) — compile-verified
//
#include <hip/hip_runtime.h>
#include <hip/hip_bf16.h>

typedef __attribute__((ext_vector_type(2))) float v2f;
typedef __attribute__((ext_vector_type(8))) float v8f;

#define E 7

// ---------------------------------------------------------------------------
// Setup kernel (1 block, 128 threads): fold all weights into a tiny param
// block in d_ws:
//   ws[0..127]  : Apad[16][8], Apad[i][l] = C[l][i] = sum_j Wq[j,l]*Wk[j,i]
//                 (zero-padded to the 16x8 WMMA A-tile)
//   ws[128..135]: colsum[i] = sum_j Wd[j,i]   (i<7, else 0)
//   ws[136]     : sumbd = sum_j bd[j]
// ---------------------------------------------------------------------------
__global__ void emotion_setup_kernel(const float* __restrict__ Wq,
                                     const float* __restrict__ Wk,
                                     const float* __restrict__ Wd,
                                     const float* __restrict__ bd,
                                     float* __restrict__ ws) {
    const int t = threadIdx.x;           // 0..127
    const int i = t >> 3;                // M row of A (output index), 0..15
    const int l = t & 7;                 // K column, 0..7
    float c = 0.0f;
    if (i < E && l < E) {
#pragma unroll
        for (int j = 0; j < E; ++j)
            c += Wq[j * E + l] * Wk[j * E + i];
    }
    ws[i * 8 + l] = c;
    if (t < 8) {
        float cs = 0.0f;
        if (t < E) {
#pragma unroll
            for (int j = 0; j < E; ++j) cs += Wd[j * E + t];
        }
        ws[128 + t] = cs;
    }
    if (t == 8) {
        float sb = 0.0f;
#pragma unroll
        for (int j = 0; j < E; ++j) sb += bd[j];
        ws[136] = sb;
    }
}

// ---------------------------------------------------------------------------
// Main streaming kernel. Each wave32 processes a 32-row tile-pair per
// iteration (grid-stride):
//   - 4x V_WMMA_F32_16X16X4_F32: two K=8 (padded from 7) matvec tiles,
//     D[i, b] = sum_l Apad[i,l] * x[b, l]  for b in [base,base+16) and
//     [base+16,base+32).
//   - Tile-0 results live in lanes 0-15 (lane n = row base+n, t[i]=acc0[i]);
//     tile-1 results are shuffled from lanes 0-15 into lanes 16-31 so ALL
//     32 lanes run the per-lane softmax epilogue on 32 distinct rows.
// A layout (16x4 f32): lanes 0-15 M=0..15; VGPR0 holds K=k0 (lanes 0-15) /
// K=k0+2 (lanes 16-31); VGPR1 holds K=k0+1 / K=k0+3. B (4x16) mirrors the
// K striping with lane = column N.
// ---------------------------------------------------------------------------
__global__ __launch_bounds__(256) void emotion_main_kernel(
    const float* __restrict__ rep,   // representation [rows, 7]
    const float* __restrict__ x,     // input_emotion  [rows, 7]
    const float* __restrict__ ws,    // param block (see setup)
    float* __restrict__ out,         // [rows, 7]
    int rows, int ntp) {             // ntp = number of 32-row tile-pairs
    const int lane = threadIdx.x & 31;
    const int m    = lane & 15;       // row-within-tile this lane serves
    const int kh   = lane >> 4;       // K-half selector for A/B striping

    // ---- A-matrix registers (uniform per kernel, loaded once) ----
    v2f A0, A1;
    A0.x = ws[m * 8 + 2 * kh];
    A0.y = ws[m * 8 + 2 * kh + 1];
    A1.x = ws[m * 8 + 4 + 2 * kh];
    A1.y = ws[m * 8 + 5 + 2 * kh];

    // ---- uniform epilogue constants (scalar loads -> SGPRs) ----
    float csum[E];
#pragma unroll
    for (int i = 0; i < E; ++i) csum[i] = ws[128 + i];
    const float sumbd = ws[136];

    const int wave   = blockIdx.x * (blockDim.x >> 5) + (threadIdx.x >> 5);
    const int nwaves = gridDim.x * (blockDim.x >> 5);

    for (int tp = wave; tp < ntp; tp += nwaves) {
        const int base = tp * 32;

        // ---- B-matrix registers for both tiles (clamped for tail) ----
        int r0 = base + m;       if (r0 >= rows) r0 = rows - 1;
        int r1 = base + 16 + m;  if (r1 >= rows) r1 = rows - 1;
        const float* x0 = x + (size_t)r0 * E;
        const float* x1 = x + (size_t)r1 * E;
        v2f B00, B01, B10, B11;
        if (kh == 0) {
            B00.x = x0[0]; B00.y = x0[1]; B01.x = x0[4]; B01.y = x0[5];
            B10.x = x1[0]; B10.y = x1[1]; B11.x = x1[4]; B11.y = x1[5];
        } else {
            B00.x = x0[2]; B00.y = x0[3]; B01.x = x0[6]; B01.y = 0.0f;
            B10.x = x1[2]; B10.y = x1[3]; B11.x = x1[6]; B11.y = 0.0f;
        }

        // prefetch next grid-stride tile-pair (global_prefetch_b8)
        {
            const long long nr = (long long)(tp + nwaves) * 32 + lane;
            if (nr < rows) {
                __builtin_prefetch(x   + (size_t)nr * E, 0, 0);
                __builtin_prefetch(rep + (size_t)nr * E, 0, 0);
            }
        }

        v8f acc0 = {};
        acc0 = __builtin_amdgcn_wmma_f32_16x16x4_f32(
            false, A0, false, B00, (short)0, acc0, false, false);
        acc0 = __builtin_amdgcn_wmma_f32_16x16x4_f32(
            false, A1, false, B01, (short)0, acc0, false, false);
        v8f acc1 = {};
        acc1 = __builtin_amdgcn_wmma_f32_16x16x4_f32(
            false, A0, false, B10, (short)0, acc1, false, false);
        acc1 = __builtin_amdgcn_wmma_f32_16x16x4_f32(
            false, A1, false, B11, (short)0, acc1, false, false);

        // ---- spread results: lane L owns row base+L for all 32 lanes ----
        float t[E];
#pragma unroll
        for (int i = 0; i < E; ++i) {
            const float sh = __shfl(acc1[i], m, 32);   // lane L<-L-16's acc1
            t[i] = (kh == 0) ? acc0[i] : sh;
        }

        // ---- full-wave per-lane epilogue ----
        const int  erow  = base + lane;
        const bool valid = erow < rows;
        const int  cr    = valid ? erow : rows - 1;
        const float* rr  = rep + (size_t)cr * E;
        float r[E], raw[E];
#pragma unroll
        for (int i = 0; i < E; ++i) r[i] = rr[i];
#pragma unroll
        for (int i = 0; i < E; ++i) raw[i] = r[i] * t[i];

        float mx = raw[0];
#pragma unroll
        for (int i = 1; i < E; ++i) mx = fmaxf(mx, raw[i]);
        float e[E], sum = 0.0f;
#pragma unroll
        for (int i = 0; i < E; ++i) {
            e[i] = __expf(raw[i] - mx);
            sum += e[i];
        }
        const float inv = __builtin_amdgcn_rcpf(sum);  // v_rcp_f32, <=1 ulp

        float o[E];
#pragma unroll
        for (int i = 0; i < E; ++i) {
            const float s  = e[i] * inv;
            const float s3 = s * s * s;
            float d = fmaf(s3, csum[i], sumbd);
            d = fminf(fmaxf(d, -1.0f), 1.0f);
            o[i] = r[i] + d;
        }
        if (valid) {
            float* op = out + (size_t)erow * E;
#pragma unroll
            for (int i = 0; i < E; ++i) op[i] = o[i];
        }
    }
}

// ---------------------------------------------------------------------------
// Launcher
// ---------------------------------------------------------------------------
extern "C" void kernel_launch(void* const* d_in, const int* in_sizes, int n_in,
                              void* d_out, int out_size, void* d_ws, size_t ws_size,
                              hipStream_t stream) {
    const float* rep = (const float*)d_in[0];   // representation [B,7]
    const float* x   = (const float*)d_in[1];   // input_emotion  [B,7]
    const float* Wq  = (const float*)d_in[2];
    const float* Wk  = (const float*)d_in[3];
    const float* Wd  = (const float*)d_in[4];
    const float* bd  = (const float*)d_in[5];
    float* ws  = (float*)d_ws;
    float* out = (float*)d_out;

    const int rows = in_sizes[0] / E;
    const int ntp  = (rows + 31) / 32;          // 32-row tile-pairs

    emotion_setup_kernel<<<1, 128, 0, stream>>>(Wq, Wk, Wd, bd, ws);

    // ~2 tile-pairs per wave: ~16K resident waves -> ~29 MB of loads in
    // flight, comfortably covering HBM latency at 23.3 TB/s.
    const int wavesPerBlock = 8;                 // 256 threads, wave32
    int targetWaves = (ntp + 1) / 2;
    int blocks = (targetWaves + wavesPerBlock - 1) / wavesPerBlock;
    if (blocks < 1) blocks = 1;

    emotion_main_kernel<<<blocks, 256, 0, stream>>>(rep, x, ws, out, rows, ntp);
}